// ELM_AE_Spectral_Pool2d_torch_2671469658356
// MI455X (gfx1250) — compile-verified
//
#include <hip/hip_runtime.h>
#include <math.h>

typedef __attribute__((ext_vector_type(2))) float v2f;
typedef __attribute__((ext_vector_type(8))) float v8f;

// Problem geometry from the reference (Q=1 hardcoded there).
constexpr int B  = 32;
constexpr int Z  = 2048;
constexpr int N  = 1024;      // W*H = 32*32
constexpr int ZC = 8;         // z-chunks for the first pass
constexpr int ZCHUNK = Z / ZC; // 256

// ---------------------------------------------------------------------------
// Kernel 1: s_part[b, zc, n] = sum_{z in chunk zc} alpha[z] * x[b, z, n]
// Lanes span n -> fully coalesced b32 loads; 8192 waves keep HBM saturated.
// ---------------------------------------------------------------------------
__global__ void k_partial(const float* __restrict__ x,
                          const float* __restrict__ alpha,
                          float* __restrict__ s_part) {
    int n  = blockIdx.x * blockDim.x + threadIdx.x;  // 0..N-1
    int zc = blockIdx.y;                             // 0..ZC-1
    int b  = blockIdx.z;                             // 0..B-1
    const float* xp = x + ((size_t)b * Z + (size_t)zc * ZCHUNK) * N + n;
    const float* ap = alpha + zc * ZCHUNK;           // uniform -> s_loads
    float acc = 0.f;
#pragma unroll 4
    for (int i = 0; i < ZCHUNK; ++i)
        acc = fmaf(ap[i], xp[(size_t)i * N], acc);
    s_part[((size_t)b * ZC + zc) * N + n] = acc;
}

// ---------------------------------------------------------------------------
// Kernel 1b: h[b,n] = sigmoid(sum_zc s_part); denom[b] = sum_n h^2
// ---------------------------------------------------------------------------
__global__ void k_hidden(const float* __restrict__ s_part,
                         float* __restrict__ h,
                         float* __restrict__ denom) {
    __shared__ float red[N];
    int b = blockIdx.x;
    int n = threadIdx.x;
    float s = 0.f;
#pragma unroll
    for (int zc = 0; zc < ZC; ++zc)
        s += s_part[((size_t)b * ZC + zc) * N + n];
    float hv = 1.f / (1.f + __expf(-s));
    h[b * N + n] = hv;
    red[n] = hv * hv;
    __syncthreads();
    for (int st = N / 2; st > 0; st >>= 1) {
        if (n < st) red[n] += red[n + st];
        __syncthreads();
    }
    if (n == 0) denom[b] = red[0];
}

// ---------------------------------------------------------------------------
// Kernel 2: beta[b,z] = (sum_n h[n] * x[b,z,n]) / denom[b] via
// V_WMMA_F32_16X16X4_F32. Per wave: one 16-row z tile, K=4 n-chunks.
//   A (16x4, MxK): lane 0-15 = row z0+lane (K=0,1 in a.x,a.y),
//                  lane 16-31 = same rows (K=2,3)        [ISA 7.12.2 layout]
//   B (4x16): h[n0+k] broadcast across all 16 columns (mirrored half-wave
//             K layout), so every D column holds the same dot product.
//   C chained across n; two accumulators shorten the WMMA dependence chain.
// ---------------------------------------------------------------------------
__global__ void k_beta(const float* __restrict__ x,
                       const float* __restrict__ h,
                       const float* __restrict__ denom,
                       float* __restrict__ out) {
    __shared__ float lh[N];
    int b    = blockIdx.x >> 4;   // 16 blocks per batch
    int zblk = blockIdx.x & 15;   // 128 z rows per block (8 waves * 16)
    int tid  = threadIdx.x;

    // Stage h[b, :] in LDS (coalesced global, broadcast ds reads later).
#pragma unroll
    for (int i = tid; i < N; i += 256)
        lh[i] = h[b * N + i];
    __syncthreads();

    int wave = tid >> 5;
    int lane = tid & 31;
    int z0   = zblk * 128 + wave * 16;
    int zrow = z0 + (lane & 15);
    int koff = (lane >> 4) << 1;  // 0 for lanes 0-15, 2 for lanes 16-31
    const float* xrow = x + ((size_t)b * Z + zrow) * N;

    v8f acc0 = {0.f, 0.f, 0.f, 0.f, 0.f, 0.f, 0.f, 0.f};
    v8f acc1 = {0.f, 0.f, 0.f, 0.f, 0.f, 0.f, 0.f, 0.f};

    for (int n0 = 0; n0 < N; n0 += 8) {
        v2f a0 = *(const v2f*)(xrow + n0 + koff);          // global b64
        v2f a1 = *(const v2f*)(xrow + n0 + 4 + koff);
        v2f b0; b0.x = lh[n0 + koff];     b0.y = lh[n0 + koff + 1];      // ds b64
        v2f b1; b1.x = lh[n0 + 4 + koff]; b1.y = lh[n0 + 4 + koff + 1];
        acc0 = __builtin_amdgcn_wmma_f32_16x16x4_f32(
                   false, a0, false, b0, (short)0, acc0, false, false);
        acc1 = __builtin_amdgcn_wmma_f32_16x16x4_f32(
                   false, a1, false, b1, (short)0, acc1, false, false);
    }

    v8f d = acc0 + acc1;
    union { v8f v; float f[8]; } u;
    u.v = d;

    // D layout: VGPR r -> M=r (lanes 0-15), M=8+r (lanes 16-31); all N
    // columns identical, so lanes 0 and 16 carry the full 16 results.
    if ((lane & 15) == 0) {
        float inv = 1.f / denom[b];
        int zout = z0 + ((lane >> 4) ? 8 : 0);
#pragma unroll
        for (int r = 0; r < 8; ++r)
            out[b * Z + zout + r] = u.f[r] * inv;
    }
}

// ---------------------------------------------------------------------------
extern "C" void kernel_launch(void* const* d_in, const int* in_sizes, int n_in,
                              void* d_out, int out_size, void* d_ws, size_t ws_size,
                              hipStream_t stream) {
    (void)in_sizes; (void)n_in; (void)out_size; (void)ws_size;
    const float* x     = (const float*)d_in[0];   // (B, Z, 32, 32) f32
    const float* alpha = (const float*)d_in[1];   // (1, Z) f32
    float* out = (float*)d_out;                   // (B, Z, 1) f32

    float* s_part = (float*)d_ws;                          // B*ZC*N floats
    float* hbuf   = s_part + (size_t)B * ZC * N;           // B*N floats
    float* denom  = hbuf + (size_t)B * N;                  // B floats

    dim3 g1(N / 256, ZC, B);
    k_partial<<<g1, 256, 0, stream>>>(x, alpha, s_part);
    k_hidden<<<B, N, 0, stream>>>(s_part, hbuf, denom);
    k_beta<<<B * 16, 256, 0, stream>>>(x, hbuf, denom, out);
}